// SS2D_73564199845999
// MI455X (gfx1250) — compile-verified
//
#include <hip/hip_runtime.h>
#include <math.h>

// Problem constants (from reference)
#define KD   4          // scan directions
#define BSZ  4
#define DI   192        // d_inner
#define NS   16         // d_state
#define RK   12         // dt_rank
#define OC   44         // RK + 2*NS
#define LL   4096       // H*W
#define NC   64         // number of scan chunks
#define CH   64         // chunk length (NC*CH == LL)

typedef float v2f __attribute__((ext_vector_type(2)));
typedef float v8f __attribute__((ext_vector_type(8)));
typedef unsigned int u32x4 __attribute__((ext_vector_type(4)));
typedef int i32x4 __attribute__((ext_vector_type(4)));
typedef int i32x8 __attribute__((ext_vector_type(8)));

// ---------------------------------------------------------------------------
// TDM: async 2D tile load global -> LDS (gfx1250 Tensor Data Mover).
// Copies tile_d1 rows x tile_d0 elements (f32), row stride = stride_elems,
// from gaddr into LDS at byte offset lds_off. Tracked by TENSORcnt.
// D# layout per CDNA5 ISA ch.8 (group0: count/lds/global/type, group1: dims).
// This toolchain's builtin takes 6 args (g0, g1, g2, g3, g4, cpol).
// ---------------------------------------------------------------------------
__device__ __forceinline__ void tdm_load_2d_f32(unsigned int lds_off,
                                                const void* gaddr,
                                                unsigned int tile_d0,
                                                unsigned int tile_d1,
                                                unsigned int stride_elems) {
    unsigned long long ga = (unsigned long long)(uintptr_t)gaddr;
    u32x4 g0;
    g0[0] = 1u;                                          // count=1 (valid user D#)
    g0[1] = lds_off;                                     // lds_addr (bytes)
    g0[2] = (unsigned int)(ga & 0xFFFFFFFFu);            // global_addr[31:0]
    g0[3] = (unsigned int)((ga >> 32) & 0x01FFFFFFu)     // global_addr[56:32]
          | (2u << 30);                                  // type=2 ("image")
    i32x8 g1;
    g1[0] = (int)(2u << 16);                             // data_size=2 -> 4 bytes
    g1[1] = (int)((tile_d0 & 0xFFFFu) << 16);            // tensor_dim0[15:0]
    g1[2] = (int)((tile_d1 & 0xFFFFu) << 16);            // tensor_dim1[15:0]
    g1[3] = (int)((tile_d0 & 0xFFFFu) << 16);            // tile_dim0
    g1[4] = (int)(tile_d1 & 0xFFFFu);                    // tile_dim1 (tile_dim2=0)
    g1[5] = (int)stride_elems;                           // tensor_dim0_stride[31:0]
    g1[6] = 0;
    g1[7] = 0;
    i32x4 gz4 = {0, 0, 0, 0};                            // groups 2/3 unused (2D)
    i32x8 gz8 = {0, 0, 0, 0, 0, 0, 0, 0};
    __builtin_amdgcn_tensor_load_to_lds(g0, g1, gz4, gz4, gz8, 0);
}

// ---------------------------------------------------------------------------
// Kernel 1: cross-scan. u_t[(k*B+b), l, d] = x[b, d, map_k(l)]   ((l,d)-major)
// ---------------------------------------------------------------------------
__global__ void build_u(const float* __restrict__ x, float* __restrict__ u_t) {
    int t = blockIdx.x * blockDim.x + threadIdx.x;      // over K*B*L*D, d fastest
    if (t >= KD * BSZ * LL * DI) return;
    int d = t % DI;
    int l = (t / DI) % LL;
    int b = (t / (DI * LL)) % BSZ;
    int k = t / (DI * LL * BSZ);
    int lp  = (k >= 2) ? (LL - 1 - l) : l;              // reverse for k=2,3
    int src = (k & 1) ? ((lp & 63) * 64 + (lp >> 6))    // col-major scan
                      : lp;                              // row-major scan
    u_t[t] = x[((size_t)b * DI + d) * LL + src];
}

// ---------------------------------------------------------------------------
// Kernel 2: projection GEMM via WMMA f32 16x16x4.
//   proj_t[l, o] = sum_d u_t[l, d] * xw[o, d]     per (k,b); o in [0,44)
// ---------------------------------------------------------------------------
__global__ void proj_wmma(const float* __restrict__ u_t,
                          const float* __restrict__ xw,
                          float* __restrict__ proj_t) {
    int job   = blockIdx.x;                    // (kb, ltile, otile), otile fastest
    int otile = job % 3;
    int ltile = (job / 3) % (LL / 16);
    int kb    = job / (3 * (LL / 16));
    int k     = kb / BSZ;

    int lane = threadIdx.x;
    int m    = lane & 15;
    int lh   = lane >> 4;

    const float* ub = u_t    + (size_t)kb * LL * DI;
    const float* wb = xw     + (size_t)k  * OC * DI;
    float*       pb = proj_t + (size_t)kb * LL * OC;

    int   lrow  = ltile * 16 + m;
    int   ocol  = otile * 16 + m;              // same lane&15 role for B operand
    float wmask = (ocol < OC) ? 1.0f : 0.0f;   // pad rows 44..47 with zeros
    int   osafe = (ocol < OC) ? ocol : 0;

    v8f acc = {};
    #pragma unroll 4
    for (int kk = 0; kk < DI; kk += 4) {
        v2f a, bm;
        const float* ap = ub + (size_t)lrow * DI + kk + 2 * lh;
        a.x = ap[0]; a.y = ap[1];
        const float* bp = wb + (size_t)osafe * DI + kk + 2 * lh;
        bm.x = bp[0] * wmask; bm.y = bp[1] * wmask;
        acc = __builtin_amdgcn_wmma_f32_16x16x4_f32(false, a, false, bm,
                                                    (short)0, acc, false, false);
    }
    if (ocol < OC) {
        #pragma unroll
        for (int j = 0; j < 8; ++j) {
            int r = ltile * 16 + j + 8 * lh;
            pb[(size_t)r * OC + ocol] = acc[j];
        }
    }
}

// ---------------------------------------------------------------------------
// Kernel 3: dt GEMM via WMMA + softplus.
//   dt_t[l, d] = softplus( sum_r proj_t[l, r] * dtw[d, r] + dtb[d] )
// ---------------------------------------------------------------------------
__global__ void dt_wmma(const float* __restrict__ proj_t,
                        const float* __restrict__ dtw,
                        const float* __restrict__ dtb,
                        float* __restrict__ dt_t) {
    int job   = blockIdx.x;                    // (kb, ltile, dtile), dtile fastest
    int dtile = job % (DI / 16);
    int ltile = (job / (DI / 16)) % (LL / 16);
    int kb    = job / ((DI / 16) * (LL / 16));
    int k     = kb / BSZ;

    int lane = threadIdx.x;
    int m    = lane & 15;
    int lh   = lane >> 4;

    const float* pb = proj_t + (size_t)kb * LL * OC;
    const float* wb = dtw    + (size_t)k  * DI * RK;
    const float* bb = dtb    + (size_t)k  * DI;
    float*       db = dt_t   + (size_t)kb * LL * DI;

    int lrow = ltile * 16 + m;
    int dcol = dtile * 16 + m;

    v8f acc = {};
    #pragma unroll
    for (int kk = 0; kk < RK; kk += 4) {       // 12 = 3 x 4
        v2f a, bm;
        const float* ap = pb + (size_t)lrow * OC + kk + 2 * lh;
        a.x = ap[0]; a.y = ap[1];
        const float* bp = wb + (size_t)dcol * RK + kk + 2 * lh;
        bm.x = bp[0]; bm.y = bp[1];
        acc = __builtin_amdgcn_wmma_f32_16x16x4_f32(false, a, false, bm,
                                                    (short)0, acc, false, false);
    }
    float bias = bb[dcol];
    #pragma unroll
    for (int j = 0; j < 8; ++j) {
        int   r  = ltile * 16 + j + 8 * lh;
        float xv = acc[j] + bias;
        // stable softplus: max(x,0) + log1p(exp(-|x|))
        float sp = fmaxf(xv, 0.0f) + log1pf(__expf(-fabsf(xv)));
        db[(size_t)r * DI + dcol] = sp;
    }
}

// ---------------------------------------------------------------------------
// Kernel 4: scan pass 1 — per-chunk end state (from h0=0) and total decay.
// Block = (kb, chunk), 192 threads (one per d). B rows staged via TDM.
// ---------------------------------------------------------------------------
__global__ void scan_pass1(const float* __restrict__ u_t,
                           const float* __restrict__ dt_t,
                           const float* __restrict__ proj_t,
                           const float* __restrict__ A_logs,
                           float* __restrict__ hend,
                           float* __restrict__ ptot) {
    __shared__ float sB[CH * NS];              // 4 KB: B rows for this chunk
    int c  = blockIdx.x % NC;
    int kb = blockIdx.x / NC;
    int k  = kb / BSZ;
    int d  = threadIdx.x;
    int l0 = c * CH;

    const float* ub = u_t    + ((size_t)kb * LL + l0) * DI;
    const float* db = dt_t   + ((size_t)kb * LL + l0) * DI;
    const float* pb = proj_t + ((size_t)kb * LL + l0) * OC;

    // Async TDM copy: CH rows x NS floats, row stride OC, global -> LDS.
    if (threadIdx.x < 32) {                    // wave 0 only (wave-uniform branch)
        unsigned int lds_off = (unsigned int)(uintptr_t)(void*)&sB[0];
        tdm_load_2d_f32(lds_off, pb + RK, NS, CH, OC);
        __builtin_amdgcn_s_wait_tensorcnt(0);
    }

    float a[NS], h[NS];
    const float* Ab = A_logs + ((size_t)k * DI + d) * NS;
    #pragma unroll
    for (int n = 0; n < NS; ++n) { a[n] = -__expf(Ab[n]); h[n] = 0.0f; }
    __syncthreads();

    float cumT = 0.0f;
    for (int i = 0; i < CH; ++i) {
        float dtv = db[(size_t)i * DI + d];
        float uv  = ub[(size_t)i * DI + d];
        float du  = dtv * uv;
        cumT += dtv;
        #pragma unroll
        for (int n = 0; n < NS; ++n) {
            float dA = __expf(dtv * a[n]);
            h[n] = dA * h[n] + du * sB[i * NS + n];
        }
    }
    size_t base = (((size_t)kb * NC + c) * DI + d) * NS;
    #pragma unroll
    for (int n = 0; n < NS; ++n) {
        hend[base + n] = h[n];
        ptot[base + n] = __expf(cumT * a[n]);  // exact product of per-step decays
    }
}

// ---------------------------------------------------------------------------
// Kernel 5: scan pass 2 — sequential combine over the 64 chunks.
// ---------------------------------------------------------------------------
__global__ void scan_pass2(const float* __restrict__ hend,
                           const float* __restrict__ ptot,
                           float* __restrict__ H0) {
    int t = blockIdx.x * blockDim.x + threadIdx.x;
    if (t >= KD * BSZ * DI * NS) return;
    int dn = t % (DI * NS);
    int kb = t / (DI * NS);
    float h = 0.0f;
    for (int c = 0; c < NC; ++c) {
        size_t idx = ((size_t)kb * NC + c) * (DI * NS) + dn;
        H0[idx] = h;
        h = ptot[idx] * h + hend[idx];
    }
}

// ---------------------------------------------------------------------------
// Kernel 6: scan pass 3 — rescan each chunk from its true init state, emit y.
// B and C rows staged via TDM (64 rows x 32 floats, stride 44).
// ---------------------------------------------------------------------------
__global__ void scan_pass3(const float* __restrict__ u_t,
                           const float* __restrict__ dt_t,
                           const float* __restrict__ proj_t,
                           const float* __restrict__ A_logs,
                           const float* __restrict__ Ds,
                           const float* __restrict__ H0,
                           float* __restrict__ y_t) {
    __shared__ float sBC[CH * 2 * NS];         // 8 KB: B and C rows per chunk
    int c  = blockIdx.x % NC;
    int kb = blockIdx.x / NC;
    int k  = kb / BSZ;
    int d  = threadIdx.x;
    int l0 = c * CH;

    const float* ub = u_t    + ((size_t)kb * LL + l0) * DI;
    const float* db = dt_t   + ((size_t)kb * LL + l0) * DI;
    const float* pb = proj_t + ((size_t)kb * LL + l0) * OC;
    float*       yb = y_t    + ((size_t)kb * LL + l0) * DI;

    if (threadIdx.x < 32) {                    // wave 0 only
        unsigned int lds_off = (unsigned int)(uintptr_t)(void*)&sBC[0];
        tdm_load_2d_f32(lds_off, pb + RK, 2 * NS, CH, OC);
        __builtin_amdgcn_s_wait_tensorcnt(0);
    }

    float a[NS], h[NS];
    const float* Ab = A_logs + ((size_t)k * DI + d) * NS;
    size_t hb = (((size_t)kb * NC + c) * DI + d) * NS;
    #pragma unroll
    for (int n = 0; n < NS; ++n) { a[n] = -__expf(Ab[n]); h[n] = H0[hb + n]; }
    float Dc = Ds[(size_t)k * DI + d];
    __syncthreads();

    for (int i = 0; i < CH; ++i) {
        float dtv = db[(size_t)i * DI + d];
        float uv  = ub[(size_t)i * DI + d];
        float du  = dtv * uv;
        float y   = Dc * uv;
        #pragma unroll
        for (int n = 0; n < NS; ++n) {
            float dA = __expf(dtv * a[n]);
            h[n] = dA * h[n] + du * sBC[i * 32 + n];
            y   += h[n] * sBC[i * 32 + 16 + n];
        }
        yb[(size_t)i * DI + d] = y;
    }
}

// ---------------------------------------------------------------------------
// Kernel 7: cross-merge. out[b,d,h,w] = y0[l] + y1[lc] + y2[L-1-l] + y3[L-1-lc]
// ---------------------------------------------------------------------------
__global__ void merge_out(const float* __restrict__ y_t, float* __restrict__ out) {
    int t = blockIdx.x * blockDim.x + threadIdx.x;     // over B*L*D, d fastest
    if (t >= BSZ * LL * DI) return;
    int d = t % DI;
    int l = (t / DI) % LL;
    int b = t / (DI * LL);
    int hh = l >> 6, ww = l & 63;
    int lc = ww * 64 + hh;

    size_t skb = (size_t)LL * DI;
    const float* y0 = y_t + ((size_t)(0 * BSZ + b)) * skb;
    const float* y1 = y_t + ((size_t)(1 * BSZ + b)) * skb;
    const float* y2 = y_t + ((size_t)(2 * BSZ + b)) * skb;
    const float* y3 = y_t + ((size_t)(3 * BSZ + b)) * skb;

    float s = y0[(size_t)l * DI + d]
            + y1[(size_t)lc * DI + d]
            + y2[(size_t)(LL - 1 - l) * DI + d]
            + y3[(size_t)(LL - 1 - lc) * DI + d];
    out[((size_t)b * DI + d) * LL + l] = s;
}

// ---------------------------------------------------------------------------
extern "C" void kernel_launch(void* const* d_in, const int* in_sizes, int n_in,
                              void* d_out, int out_size, void* d_ws, size_t ws_size,
                              hipStream_t stream) {
    (void)in_sizes; (void)n_in; (void)out_size; (void)ws_size;
    const float* x    = (const float*)d_in[0];   // (B, D, H, W)
    const float* xw   = (const float*)d_in[1];   // (K, 44, 192)
    const float* dtw  = (const float*)d_in[2];   // (K, 192, 12)
    const float* dtb  = (const float*)d_in[3];   // (K, 192)
    const float* Alog = (const float*)d_in[4];   // (K, 192, 16)
    const float* Ds   = (const float*)d_in[5];   // (K, 192)
    float* out = (float*)d_out;

    // workspace layout (floats)
    const size_t SZ_UD = (size_t)KD * BSZ * LL * DI;        // 12,582,912
    const size_t SZ_P  = (size_t)KD * BSZ * LL * OC;        //  2,883,584
    const size_t SZ_S  = (size_t)KD * BSZ * NC * DI * NS;   //  3,145,728
    float* ws     = (float*)d_ws;
    float* u_t    = ws;
    float* dt_t   = u_t  + SZ_UD;
    float* y_t    = dt_t + SZ_UD;
    float* proj_t = y_t  + SZ_UD;
    float* hend   = proj_t + SZ_P;
    float* ptot   = hend   + SZ_S;
    float* H0     = ptot   + SZ_S;

    // 1) cross-scan
    {
        int total = KD * BSZ * LL * DI;
        build_u<<<(total + 255) / 256, 256, 0, stream>>>(x, u_t);
    }
    // 2) x_proj GEMM (WMMA): one wave per 16x16 tile
    {
        int grid = KD * BSZ * (LL / 16) * 3;
        proj_wmma<<<grid, 32, 0, stream>>>(u_t, xw, proj_t);
    }
    // 3) dt GEMM (WMMA) + softplus
    {
        int grid = KD * BSZ * (LL / 16) * (DI / 16);
        dt_wmma<<<grid, 32, 0, stream>>>(proj_t, dtw, dtb, dt_t);
    }
    // 4-6) chunked selective scan
    scan_pass1<<<KD * BSZ * NC, DI, 0, stream>>>(u_t, dt_t, proj_t, Alog, hend, ptot);
    {
        int total = KD * BSZ * DI * NS;
        scan_pass2<<<(total + 255) / 256, 256, 0, stream>>>(hend, ptot, H0);
    }
    scan_pass3<<<KD * BSZ * NC, DI, 0, stream>>>(u_t, dt_t, proj_t, Alog, Ds, H0, y_t);
    // 7) cross-merge
    {
        int total = BSZ * LL * DI;
        merge_out<<<(total + 255) / 256, 256, 0, stream>>>(y_t, out);
    }
}